// HGT_1864015807111
// MI455X (gfx1250) — compile-verified
//
#include <hip/hip_runtime.h>

// ---------------------------------------------------------------------------
// HGT-style graph transformer layer for MI455X (gfx1250, wave32, WMMA)
//   N=100000 nodes, E=1.6M edges, IN=256, OUT=128, H=8, DK=16
// Dense GEMMs use V_WMMA_F32_16X16X4_F32 with W staged in LDS (pair-
// interleaved so each B fragment is one ds_load_b64). Edge phase is
// L2-resident gathers + f32 atomics (the real bottleneck, ~2.6GB traffic).
// ---------------------------------------------------------------------------

#define OUTD 128
#define HEADS 8
#define DK 16

typedef __attribute__((ext_vector_type(2))) float v2f;
typedef __attribute__((ext_vector_type(8))) float v8f;

// ---------------------------------------------------------------------------
// C[M x 128] = A[M x K] @ W[K x 128] + bias[128]   (K multiple of 4, K<=256)
// Block = 256 threads = 8 waves; wave w computes rows [blk*128 + w*16, +16).
// Each wave holds eight 16x16 f32 accumulators covering all 128 columns.
// W is staged once per block into LDS as float2 pairs of adjacent K-rows.
// Safe to run in place (C == A): each wave reads only its own 16 A-rows and
// stores them only after all WMMAs (data dependency enforces order).
// OOB rows: clamped address, garbage computed but never stored (row m of D
// depends only on row m of A) -> no EXEC divergence around WMMA.
// ---------------------------------------------------------------------------
__global__ __launch_bounds__(256) void gemm_bias_wmma_f32(
    const float* __restrict__ A, const float* __restrict__ W,
    const float* __restrict__ bias, float* __restrict__ C,
    int M, int K)
{
    // Wl[i][c] = (W[2i][c], W[2i+1][c]); K<=256 -> up to 128 pair-rows
    __shared__ v2f Wl[128][OUTD];

    const int tid  = threadIdx.x;
    const int wave = tid >> 5;
    const int lane = tid & 31;
    const int half = lane >> 4;   // 0: lanes 0-15, 1: lanes 16-31
    const int lr   = lane & 15;
    const int rowBase = blockIdx.x * 128 + wave * 16;

    // ---- stage W into LDS (pair-interleaved) ----
    {
        const int pairs = (K >> 1) * OUTD;       // number of float2 elements
        for (int idx = tid; idx < pairs; idx += 256) {
            const int i = idx >> 7;              // pair-row (k/2)
            const int c = idx & (OUTD - 1);      // column
            v2f w;
            w.x = W[(long)(2 * i)     * OUTD + c];
            w.y = W[(long)(2 * i + 1) * OUTD + c];
            Wl[i][c] = w;
        }
    }
    __syncthreads();

    v8f acc[8] = {};

    // A fragment rows: lanes 0-15 and 16-31 both map M = lr (ISA 16x4 layout)
    const int arow = rowBase + lr;
    const int arow_c = arow < M ? arow : (M - 1);   // clamp: no divergence
    const float* Arow = A + (long)arow_c * K;

#pragma unroll 2
    for (int k = 0; k < K; k += 4) {
        const int ka = k + 2 * half;          // half 0 -> K{0,1}, half 1 -> K{2,3}
        const v2f af = *(const v2f*)(Arow + ka);   // global_load_b64
        const int pr = ka >> 1;               // pair-row in LDS
#pragma unroll
        for (int c = 0; c < 8; ++c) {
            const v2f bf = Wl[pr][c * 16 + lr];    // ds_load_b64
            acc[c] = __builtin_amdgcn_wmma_f32_16x16x4_f32(
                false, af, false, bf, (short)0, acc[c], false, false);
        }
    }

    // C/D layout: VGPR r -> M = r (lanes 0-15) / r+8 (lanes 16-31), N = lane
#pragma unroll
    for (int c = 0; c < 8; ++c) {
        const float bb = bias[c * 16 + lr];
#pragma unroll
        for (int r = 0; r < 8; ++r) {
            const int m = rowBase + r + 8 * half;
            if (m < M)
                C[(long)m * OUTD + c * 16 + lr] = acc[c][r] + bb;
        }
    }
}

// ---------------------------------------------------------------------------
// Edge-phase helpers: monotone encoding of f32 so unsigned atomicMax == fmax
// ---------------------------------------------------------------------------
__device__ __forceinline__ unsigned enc_f32(float f) {
    unsigned u = __float_as_uint(f);
    return (u & 0x80000000u) ? ~u : (u | 0x80000000u);
}
__device__ __forceinline__ float dec_f32(unsigned u) {
    u = (u & 0x80000000u) ? (u & 0x7FFFFFFFu) : ~u;
    return __uint_as_float(u);
}

__global__ __launch_bounds__(256) void zero_f32(float* __restrict__ p, long n) {
    long i = (long)blockIdx.x * blockDim.x + threadIdx.x;
    long stride = (long)gridDim.x * blockDim.x;
    for (; i < n; i += stride) p[i] = 0.0f;
}

// SDDMM: s[e,h] = dot(k2[src[e],h,:], q[dst[e],h,:]) / sqrt(DK)
__global__ __launch_bounds__(256) void sddmm_kernel(
    const float* __restrict__ K2, const float* __restrict__ Q,
    const int* __restrict__ src, const int* __restrict__ dst,
    float* __restrict__ S, int E)
{
    int t = blockIdx.x * blockDim.x + threadIdx.x;
    if (t >= E * HEADS) return;
    int e = t >> 3, hh = t & 7;
    int sn = src[e], dn = dst[e];
    const float4* kp = (const float4*)(K2 + (long)sn * OUTD + hh * DK);
    const float4* qp = (const float4*)(Q  + (long)dn * OUTD + hh * DK);
    float acc = 0.0f;
#pragma unroll
    for (int i = 0; i < 4; ++i) {
        float4 a = kp[i], b = qp[i];
        acc += a.x * b.x + a.y * b.y + a.z * b.z + a.w * b.w;
    }
    S[t] = acc * 0.25f;   // 1/sqrt(16)
}

// segment-max over incoming edges of dst (encoded unsigned atomic max)
__global__ __launch_bounds__(256) void edge_max_kernel(
    const float* __restrict__ S, const int* __restrict__ dst,
    unsigned* __restrict__ Menc, int E)
{
    int t = blockIdx.x * blockDim.x + threadIdx.x;
    if (t >= E * HEADS) return;
    int e = t >> 3, hh = t & 7;
    atomicMax(&Menc[(long)dst[e] * HEADS + hh], enc_f32(S[t]));
}

// e = exp(s - m[dst]); z[dst] += e; store e in place of s
__global__ __launch_bounds__(256) void edge_expsum_kernel(
    float* __restrict__ S, const int* __restrict__ dst,
    const unsigned* __restrict__ Menc, float* __restrict__ Z, int E)
{
    int t = blockIdx.x * blockDim.x + threadIdx.x;
    if (t >= E * HEADS) return;
    int e = t >> 3, hh = t & 7;
    long di = (long)dst[e] * HEADS + hh;
    float m  = dec_f32(Menc[di]);
    float ev = __expf(S[t] - m);
    S[t] = ev;
    atomicAdd(&Z[di], ev);
}

// SPMM: agg[dst,h,:] += v2[src,h,:] * (e / z[dst,h])
__global__ __launch_bounds__(256) void spmm_kernel(
    const float* __restrict__ S, const float* __restrict__ V2,
    const int* __restrict__ src, const int* __restrict__ dst,
    const float* __restrict__ Z, float* __restrict__ AGG, int E)
{
    int t = blockIdx.x * blockDim.x + threadIdx.x;
    if (t >= E * HEADS) return;
    int e = t >> 3, hh = t & 7;
    int sn = src[e], dn = dst[e];
    float attn = S[t] / Z[(long)dn * HEADS + hh];
    const float4* vp = (const float4*)(V2 + (long)sn * OUTD + hh * DK);
    float* ap = AGG + (long)dn * OUTD + hh * DK;
#pragma unroll
    for (int i = 0; i < 4; ++i) {
        float4 v = vp[i];
        atomicAdd(&ap[4 * i + 0], v.x * attn);
        atomicAdd(&ap[4 * i + 1], v.y * attn);
        atomicAdd(&ap[4 * i + 2], v.z * attn);
        atomicAdd(&ap[4 * i + 3], v.w * attn);
    }
}

// ---------------------------------------------------------------------------
// host-side orchestration
// ---------------------------------------------------------------------------
extern "C" void kernel_launch(void* const* d_in, const int* in_sizes, int n_in,
                              void* d_out, int out_size, void* d_ws, size_t ws_size,
                              hipStream_t stream)
{
    const float* h     = (const float*)d_in[0];
    const float* Wq    = (const float*)d_in[1];
    const float* bq    = (const float*)d_in[2];
    const float* Wk    = (const float*)d_in[3];
    const float* bk    = (const float*)d_in[4];
    const float* Wv    = (const float*)d_in[5];
    const float* bv    = (const float*)d_in[6];
    const float* Wmsg  = (const float*)d_in[7];
    const float* bmsg  = (const float*)d_in[8];
    const float* Wattn = (const float*)d_in[9];
    const float* battn = (const float*)d_in[10];
    const float* Wa    = (const float*)d_in[11];
    const float* ba    = (const float*)d_in[12];
    const int*   src   = (const int*)d_in[13];
    const int*   dst   = (const int*)d_in[14];
    float* out = (float*)d_out;

    const int IN = 256;
    const int N  = in_sizes[0] / IN;       // 100000
    const int E  = in_sizes[13];           // 1600000

    // workspace carve-up (floats)
    float* Q    = (float*)d_ws;            // N*128  (aliased as AGG later)
    float* Kb   = Q  + (long)N * OUTD;     // N*128  (stage-2 K in place)
    float* Vb   = Kb + (long)N * OUTD;     // N*128  (stage-2 V in place)
    float* S    = Vb + (long)N * OUTD;     // E*8
    unsigned* Menc = (unsigned*)(S + (long)E * HEADS); // N*8
    float* Z    = (float*)(Menc + (long)N * HEADS);    // N*8
    float* AGG  = Q;                       // reuse: Q dead after SDDMM

    const int gGemm = (N + 127) / 128;
    const long tEdge = (long)E * HEADS;
    const int gEdge = (int)((tEdge + 255) / 256);

    // stage 1: three projections of h (K = 256)
    gemm_bias_wmma_f32<<<gGemm, 256, 0, stream>>>(h, Wq, bq, Q,  N, IN);
    gemm_bias_wmma_f32<<<gGemm, 256, 0, stream>>>(h, Wk, bk, Kb, N, IN);
    gemm_bias_wmma_f32<<<gGemm, 256, 0, stream>>>(h, Wv, bv, Vb, N, IN);

    // stage 2: K = Kb@Wattn + battn ; V = Vb@Wmsg + bmsg (in place, K = 128)
    gemm_bias_wmma_f32<<<gGemm, 256, 0, stream>>>(Kb, Wattn, battn, Kb, N, OUTD);
    gemm_bias_wmma_f32<<<gGemm, 256, 0, stream>>>(Vb, Wmsg,  bmsg,  Vb, N, OUTD);

    // zero Menc + Z (adjacent, N*16 floats; bit pattern 0 == encoded -inf)
    zero_f32<<<1024, 256, 0, stream>>>((float*)Menc, (long)N * HEADS * 2);

    // SDDMM -> S  (consumes Q)
    sddmm_kernel<<<gEdge, 256, 0, stream>>>(Kb, Q, src, dst, S, E);

    // AGG aliases Q; zero it only after SDDMM has read Q (stream-ordered)
    zero_f32<<<1024, 256, 0, stream>>>(AGG, (long)N * OUTD);

    // edge softmax: max, then exp+sum (e stored over s)
    edge_max_kernel   <<<gEdge, 256, 0, stream>>>(S, dst, Menc, E);
    edge_expsum_kernel<<<gEdge, 256, 0, stream>>>(S, dst, Menc, Z, E);

    // SPMM: weighted aggregation into AGG
    spmm_kernel<<<gEdge, 256, 0, stream>>>(S, Vb, src, dst, Z, AGG, E);

    // output projection: out = AGG @ Wa + ba
    gemm_bias_wmma_f32<<<gGemm, 256, 0, stream>>>(AGG, Wa, ba, out, N, OUTD);
}